// CapsuleRoutingPooling_48515950576313
// MI455X (gfx1250) — compile-verified
//
#include <hip/hip_runtime.h>
#include <hip/hip_bf16.h>
#include <stdint.h>

// B=16, C=64, H=64, W=64, D=16, k=2 (fixed by harness).
// Routing collapses (softmax over singleton axis) => out = squash(2x2 sum-pool).
// 320 MB of traffic @ 23.3 TB/s => ~13.7 us roofline; copy via TDM, compute trivial.

typedef float    v4f   __attribute__((ext_vector_type(4)));
typedef uint32_t u32x4 __attribute__((ext_vector_type(4)));
typedef uint32_t u32x8 __attribute__((ext_vector_type(8)));

__global__ __launch_bounds__(256) void caps_pool_kernel(const float* __restrict__ in,
                                                        float* __restrict__ out) {
    // 72 KB: 64 KB payload + 16B TDM pad per 128B row (bank-conflict mitigation).
    __shared__ v4f smem[4608];

    const uint32_t b     = blockIdx.x;   // 0..4095
    const uint32_t t     = threadIdx.x;  // 0..255
    const uint32_t bc    = b >> 2;       // 0..1023  (b*C + c)
    const uint32_t nhBlk = b & 3u;       // group of 8 nh-rows

    // Block input slab: 16 contiguous rows (h = 16*nhBlk..+15) of image bc = 64 KB.
    const uint64_t inBaseF = (uint64_t)bc * 65536u + (uint64_t)nhBlk * 16384u;

    // ---- TDM: one DMA per block, Global -> LDS with 16B pad every 128B -------
    if (t < 32u) {                        // wave 0 only (wave-uniform branch)
        const uint64_t ga  = (uint64_t)(uintptr_t)(in + inBaseF);
        const uint32_t lds = (uint32_t)(uintptr_t)(&smem[0]);

        u32x4 g0;
        g0[0] = 1u;                                   // count=1, user mode
        g0[1] = lds;                                  // lds_addr
        g0[2] = (uint32_t)ga;                         // global_addr[31:0]
        g0[3] = (uint32_t)((ga >> 32) & 0x01FFFFFFu)  // global_addr[56:32]
              | 0x80000000u;                          // type=2 ("image")

        u32x8 g1;
        g1[0] = (2u << 16)    // data_size = 4 bytes
              | (1u << 20)    // pad_enable
              | (4u << 22)    // pad_interval: code 4 = 32 dwords = 128 B
              | (3u << 25);   // pad_amount:   code 3 = 4 dwords  = 16 B
        g1[1] = 0x40000000u;  // tensor_dim0 = 16384 (low16 in bits[31:16])
        g1[2] = 0x00010000u;  // tensor_dim0 hi = 0; tensor_dim1 = 1 (bits[31:16])
        g1[3] = 0x40000000u;  // tensor_dim1 hi = 0; tile_dim0 = 16384 (bits[31:16])
        g1[4] = 0u;           // tile_dim1 = 0 (unused), tile_dim2 = 0 (unused)
        g1[5] = 16384u;       // tensor_dim0_stride[31:0]
        g1[6] = 0u;           // stride hi / tensor_dim1_stride lo
        g1[7] = 0u;

        asm volatile("tensor_load_to_lds %0, %1 th:TH_LOAD_NT"
                     :: "s"(g0), "s"(g1)
                     : "memory");
        __builtin_amdgcn_s_wait_tensorcnt(0);
    }
    __syncthreads();

    // ---- compute: one output vector per thread ------------------------------
    // Padded LDS layout (v4f units): chunk(task,r,w,c) at
    //   task*576 + r*288 + w... (w=2nw+dx inside 144B-strided pairs):
    //   task*576 + r*288 + nw*9 + dx*4 + c
    const uint32_t j    = t >> 5;        // task 0..7  (one task per wave)
    const uint32_t nw   = t & 31u;
    const uint32_t base = j * 576u + nw * 9u;

    v4f acc[4];
    #pragma unroll
    for (int c = 0; c < 4; ++c) acc[c] = (v4f){0.f, 0.f, 0.f, 0.f};

    #pragma unroll
    for (int r = 0; r < 2; ++r) {
        #pragma unroll
        for (int dx = 0; dx < 2; ++dx) {
            uint32_t idx = base + (uint32_t)r * 288u + (uint32_t)dx * 4u;
            #pragma unroll
            for (int c = 0; c < 4; ++c)
                acc[c] += smem[idx + (uint32_t)c];     // ds_load_b128, <=2-way conflicts
        }
    }

    // squash: v = sq/(1+sq) * s/(sqrt(sq)+1e-8)
    v4f d2 = acc[0]*acc[0] + acc[1]*acc[1] + acc[2]*acc[2] + acc[3]*acc[3];
    float sq = d2.x + d2.y + d2.z + d2.w;
    float scale = sq / ((1.0f + sq) * (sqrtf(sq) + 1e-8f));

    v4f* outp = reinterpret_cast<v4f*>(out);
    uint64_t o4 = (uint64_t)bc * 4096u
                + (uint64_t)(nhBlk * 8u + j) * 128u
                + (uint64_t)nw * 4u;
    #pragma unroll
    for (int c = 0; c < 4; ++c) {
        v4f rv = acc[c] * scale;
        __builtin_nontemporal_store(rv, &outp[o4 + (uint32_t)c]);  // th:TH_STORE_NT
    }
}

extern "C" void kernel_launch(void* const* d_in, const int* in_sizes, int n_in,
                              void* d_out, int out_size, void* d_ws, size_t ws_size,
                              hipStream_t stream) {
    (void)in_sizes; (void)n_in; (void)d_ws; (void)ws_size; (void)out_size;
    const float* in = (const float*)d_in[0];
    float* out = (float*)d_out;
    // 4096 blocks (1024 bc-slices x 4 row-groups) x 256 threads (8 wave32).
    caps_pool_kernel<<<dim3(4096), dim3(256), 0, stream>>>(in, out);
}